// gPool_39865886442140
// MI455X (gfx1250) — compile-verified
//
#include <hip/hip_runtime.h>
#include <hip/hip_bf16.h>

// gPool: scores = x@w + b ; top-k(4096) per batch ; indices sorted ascending ;
// gather pooled rows. B=16, N=32768, C=256, K=4096.
//
// Pipeline (all on `stream`):
//   1) score_kernel   : WMMA f32 16x16x4 GEMV -> scores[B][N]   (d_ws)
//   2) select_kernel  : per-batch MSD radix select -> threshold T + tie budget
//   3) compact_kernel : stable index-ordered compaction -> indices (ascending)
//   4) gather_kernel  : copy selected rows (float4)
//
// d_out layout: [B*K*C] pooled floats, then [B*K] indices (as float values).

#define GB 16
#define GN 32768
#define GC 256
#define GK 4096

typedef __attribute__((ext_vector_type(2))) float v2f;
typedef __attribute__((ext_vector_type(8))) float v8f;

// ---------------------------------------------------------------- scores ----
// One wave (32 lanes) computes scores for a 16-node tile using
// V_WMMA_F32_16X16X4_F32, accumulating over C=256 in K=4 chunks.
// A layout (32-bit A 16x4): lanes 0-15 -> M=0..15, K=kOff+{0,1} in the two
// VGPRs; lanes 16-31 -> same M, K=kOff+{2,3}. B is w-chunk broadcast to all 16
// columns, so D[:,j] is identical for all j; lanes 0 and 16 each hold 8 of the
// 16 scores in their accumulator VGPRs (C/D layout: lane L, vgpr r -> M=r+8*(L/16)).
__global__ __launch_bounds__(256) void score_kernel(const float* __restrict__ x,
                                                    const float* __restrict__ w,
                                                    const float* __restrict__ bias,
                                                    float* __restrict__ scores) {
    __shared__ float sw[GC];
    const int tid = threadIdx.x;
    sw[tid] = w[tid];                 // 256 threads load 256 weights
    __syncthreads();

    const int lane = tid & 31;
    const int wv   = tid >> 5;        // 8 waves per block
    const int tile = blockIdx.x * 8 + wv;             // global 16-node tile id
    const int tilesPerBatch = GN / 16;                // 2048
    const int b        = tile / tilesPerBatch;
    const int nodeBase = (tile % tilesPerBatch) * 16;

    const int m    = lane & 15;       // node within tile
    const int half = lane >> 4;       // 0 or 1
    const int kOff = half * 2;        // K sub-offset within 4-chunk

    const float* __restrict__ row = x + ((size_t)b * GN + nodeBase + m) * GC;

    v8f acc = {};
    #pragma unroll 4
    for (int c = 0; c < GC; c += 4) {
        v2f a, bb;
        a.x  = row[c + kOff];
        a.y  = row[c + kOff + 1];
        bb.x = sw[c + kOff];
        bb.y = sw[c + kOff + 1];
        acc = __builtin_amdgcn_wmma_f32_16x16x4_f32(
            /*neg_a=*/false, a, /*neg_b=*/false, bb,
            /*c_mod=*/(short)0, acc, /*reuse_a=*/false, /*reuse_b=*/false);
    }

    const float bs = bias[0];
    if (m == 0) {   // lanes 0 and 16 hold column N=0 of D = the 16 scores
        float* sOut = scores + (size_t)b * GN + nodeBase + half * 8;
        #pragma unroll
        for (int r = 0; r < 8; ++r) sOut[r] = acc[r] + bs;
    }
}

// ------------------------------------------------------------- selection ----
__device__ __forceinline__ unsigned orderedU(float f) {
    unsigned u = __float_as_uint(f);
    return u ^ ((u >> 31) ? 0xFFFFFFFFu : 0x80000000u);  // larger f -> larger u
}

// One 256-thread workgroup per batch. 4-pass MSD radix select (8 bits/pass)
// over ordered-uint keys. Output per batch: params[2b]=T (ordered uint of the
// k-th largest score), params[2b+1]=nTies (# of ==T elements to take, smallest
// indices first) -- matches jax.lax.top_k tie-breaking.
__global__ __launch_bounds__(256) void select_kernel(const float* __restrict__ scores,
                                                     unsigned* __restrict__ params) {
    __shared__ unsigned hist[256];
    __shared__ unsigned sPrefix, sKRem;
    const int b = blockIdx.x, tid = threadIdx.x;
    if (tid == 0) { sPrefix = 0u; sKRem = GK; }
    const float* __restrict__ s = scores + (size_t)b * GN;

    for (int pass = 0; pass < 4; ++pass) {
        const int shift = 24 - pass * 8;
        hist[tid] = 0u;
        __syncthreads();
        const unsigned prefix = sPrefix;
        for (int i = tid; i < GN; i += 256) {
            const unsigned u = orderedU(s[i]);
            const bool match = (pass == 0) || (((u ^ prefix) >> (shift + 8)) == 0u);
            if (match) atomicAdd(&hist[(u >> shift) & 255u], 1u);
        }
        __syncthreads();
        if (tid == 0) {
            unsigned kRem = sKRem, cum = 0u;
            int d = 255;
            for (; d > 0; --d) {
                if (cum + hist[d] >= kRem) break;
                cum += hist[d];
            }
            sKRem   = kRem - cum;                       // rank within this digit
            sPrefix = prefix | ((unsigned)d << shift);
        }
        __syncthreads();
    }
    if (tid == 0) { params[2 * b] = sPrefix; params[2 * b + 1] = sKRem; }
}

// One 1024-thread (32-wave) workgroup per batch. Stable (index-ordered)
// compaction: selected = (u > T) || (u == T && eqRank < nTies). Wave32 ballot
// prefix + LDS cross-wave scan. Emits ascending indices as floats into d_out.
__global__ __launch_bounds__(1024) void compact_kernel(const float* __restrict__ scores,
                                                       const unsigned* __restrict__ params,
                                                       float* __restrict__ outIdxF) {
    __shared__ unsigned wTotE[32], wTotS[32], wBaseE[32], wBaseS[32];
    __shared__ unsigned sBaseE, sBaseS, chunkTotE, chunkTotS;
    const int b = blockIdx.x, tid = threadIdx.x;
    const int lane = tid & 31, wv = tid >> 5;
    const unsigned T = params[2 * b];
    const unsigned nTies = params[2 * b + 1];
    if (tid == 0) { sBaseE = 0u; sBaseS = 0u; }
    __syncthreads();

    const float* __restrict__ s = scores + (size_t)b * GN;
    const unsigned laneMaskLt = (1u << lane) - 1u;

    for (int base = 0; base < GN; base += 1024) {
        const int i = base + tid;
        const unsigned u = orderedU(s[i]);
        const bool fE = (u == T);
        const bool fG = (u > T);

        // --- scan of equal flags -> global tie rank
        const unsigned balE = (unsigned)__ballot(fE);
        if (lane == 0) wTotE[wv] = (unsigned)__popc(balE);
        __syncthreads();
        if (tid == 0) {
            unsigned run = 0u;
            for (int j = 0; j < 32; ++j) { wBaseE[j] = run; run += wTotE[j]; }
            chunkTotE = run;
        }
        __syncthreads();
        const unsigned eqRank = sBaseE + wBaseE[wv] + (unsigned)__popc(balE & laneMaskLt);

        const bool sel = fG || (fE && eqRank < nTies);

        // --- scan of selected flags -> output position (ascending index order)
        const unsigned balS = (unsigned)__ballot(sel);
        if (lane == 0) wTotS[wv] = (unsigned)__popc(balS);
        __syncthreads();
        if (tid == 0) {
            unsigned run = 0u;
            for (int j = 0; j < 32; ++j) { wBaseS[j] = run; run += wTotS[j]; }
            chunkTotS = run;
        }
        __syncthreads();
        const unsigned pos = sBaseS + wBaseS[wv] + (unsigned)__popc(balS & laneMaskLt);

        if (sel) outIdxF[(size_t)b * GK + pos] = (float)i;
        __syncthreads();
        if (tid == 0) { sBaseE += chunkTotE; sBaseS += chunkTotS; }
        __syncthreads();
    }
}

// ---------------------------------------------------------------- gather ----
// 64 threads copy one 256-float row with float4 (1 KB per row). 4 rows/block.
__global__ __launch_bounds__(256) void gather_kernel(const float* __restrict__ x,
                                                     const float* __restrict__ idxF,
                                                     float* __restrict__ out) {
    const int t = threadIdx.x;
    const int rowInBlk = t >> 6;        // 0..3
    const int l64 = t & 63;             // 0..63
    const size_t r = (size_t)blockIdx.x * 4 + rowInBlk;   // 0 .. B*K-1
    const int b = (int)(r / GK);
    const int j = (int)(r % GK);
    const int idx = (int)idxF[(size_t)b * GK + j];
    const float4* __restrict__ src =
        (const float4*)(x + ((size_t)b * GN + idx) * GC);
    float4* __restrict__ dst = (float4*)(out + ((size_t)b * GK + j) * GC);
    dst[l64] = src[l64];
}

extern "C" void kernel_launch(void* const* d_in, const int* in_sizes, int n_in,
                              void* d_out, int out_size, void* d_ws, size_t ws_size,
                              hipStream_t stream) {
    (void)in_sizes; (void)n_in; (void)out_size; (void)ws_size;
    const float* x    = (const float*)d_in[0];
    const float* w    = (const float*)d_in[1];
    const float* bias = (const float*)d_in[2];
    float* out = (float*)d_out;

    float*    scores = (float*)d_ws;                                   // B*N f32 = 2 MiB
    unsigned* params = (unsigned*)((char*)d_ws + (size_t)GB * GN * sizeof(float)); // 32 u32
    float*    idxF   = out + (size_t)GB * GK * GC;                     // 2nd output region

    // 1) scores via WMMA: 8 waves/block, 16 nodes/wave -> B*N/128 blocks
    score_kernel<<<(GB * GN) / 128, 256, 0, stream>>>(x, w, bias, scores);
    // 2) per-batch radix select of k-th largest + tie budget
    select_kernel<<<GB, 256, 0, stream>>>(scores, params);
    // 3) stable compaction -> ascending indices (written as float into d_out)
    compact_kernel<<<GB, 1024, 0, stream>>>(scores, params, idxF);
    // 4) gather pooled rows
    gather_kernel<<<(GB * GK) / 4, 256, 0, stream>>>(x, idxF, out);
}